// Attention_3908420239434
// MI455X (gfx1250) — compile-verified
//
#include <hip/hip_runtime.h>
#include <hip/hip_bf16.h>

typedef __attribute__((ext_vector_type(2))) float v2f;
typedef __attribute__((ext_vector_type(4))) float v4f;
typedef __attribute__((ext_vector_type(8))) float v8f;

#define SP    8192
#define IN_F  256
#define HID   64
#define NEG_SLOPE 0.2f
#define EPS_D 1e-12f

// ---------------------------------------------------------------------------
// Kernel 1: F = leaky_relu(X @ W)   X:[8192,256] W:[256,64] F:[8192,64]
// One wave per 16x16 tile of F; K loop in steps of 4 via V_WMMA_F32_16X16X4_F32.
// A fragment: lane holds A[M=lane%16][K = 2*(lane/16)+j]  (v2f, contiguous)
// B fragment: lane holds W[K = 2*(lane/16)+j][N = lane%16] (two strided loads)
// C/D: vgpr r, lane L -> row r+8*(L/16), col L%16
// blockIdx.y selects (input1->F1) vs (input2->F2).
// ---------------------------------------------------------------------------
__global__ void proj_wmma_kernel(const float* __restrict__ X1,
                                 const float* __restrict__ X2,
                                 const float* __restrict__ W,
                                 float* __restrict__ F1,
                                 float* __restrict__ F2) {
    const float* X = (blockIdx.y == 0) ? X1 : X2;
    float*       F = (blockIdx.y == 0) ? F1 : F2;

    const int wave  = threadIdx.x >> 5;
    const int lane  = threadIdx.x & 31;
    const int hf    = lane >> 4;     // 0/1 : which K-half of the fragment
    const int l16   = lane & 15;

    const int tile = blockIdx.x * 8 + wave;   // 0..2047
    const int tN   = tile & 3;                // HID/16 = 4 tiles wide
    const int tM   = tile >> 2;               // 0..511

    const float* xrow = X + (size_t)(tM * 16 + l16) * IN_F + 2 * hf;
    const float* wcol = W + (size_t)(2 * hf) * HID + tN * 16 + l16;

    v8f c = {};
    #pragma unroll 8
    for (int kb = 0; kb < IN_F; kb += 4) {
        v2f a = *(const v2f*)(xrow + kb);
        v2f b;
        b.x = wcol[(size_t)kb * HID];
        b.y = wcol[(size_t)kb * HID + HID];
        c = __builtin_amdgcn_wmma_f32_16x16x4_f32(false, a, false, b,
                                                  (short)0, c, false, false);
    }

    float* frow = F + (size_t)(tM * 16 + 8 * hf) * HID + tN * 16 + l16;
    #pragma unroll
    for (int r = 0; r < 8; ++r) {
        float v = c[r];
        v = (v >= 0.0f) ? v : NEG_SLOPE * v;
        frow[(size_t)r * HID] = v;
    }
}

// ---------------------------------------------------------------------------
// Kernel 2: per-row sum of squares for F1 and F2 (one thread per row).
// ---------------------------------------------------------------------------
__global__ void rowsq_kernel(const float* __restrict__ F1,
                             const float* __restrict__ F2,
                             float* __restrict__ sq1,
                             float* __restrict__ sq2) {
    int i = blockIdx.x * blockDim.x + threadIdx.x;   // 0..16383
    const float* f = (i < SP) ? (F1 + (size_t)i * HID)
                              : (F2 + (size_t)(i - SP) * HID);
    float s = 0.0f;
    #pragma unroll
    for (int j = 0; j < HID; j += 4) {
        v4f v = *(const v4f*)(f + j);
        s += v.x * v.x + v.y * v.y + v.z * v.z + v.w * v.w;
    }
    if (i < SP) sq1[i] = s; else sq2[i - SP] = s;
}

__device__ __forceinline__ float sig_dist(float dot, float sm, float sn) {
    float d2   = fmaxf(sm + sn - 2.0f * dot, 0.0f);
    float dist = sqrtf(d2 + EPS_D);
    return 1.0f / (1.0f + __expf(dist));       // sigmoid(-dist)
}

// ---------------------------------------------------------------------------
// Kernel 3: out[m,n] = sigmoid(-sqrt(max(sq1[m]+sq2[n]-2*f1[m].f2[n],0)+eps))
// Each wave computes a 32x32 output block: 2x2 register tiles, 4 v8f accums,
// 16 K-steps x 4 WMMAs, each A/B fragment reused twice.
// Block = 256 threads = 8 waves arranged 2(M) x 4(N) -> 64x128 per workgroup.
// Output streamed with non-temporal stores (256 MB > L2, never re-read).
// ---------------------------------------------------------------------------
__global__ void dist_wmma_kernel(const float* __restrict__ F1,
                                 const float* __restrict__ F2,
                                 const float* __restrict__ sq1,
                                 const float* __restrict__ sq2,
                                 float* __restrict__ out) {
    const int wave = threadIdx.x >> 5;   // 0..7
    const int lane = threadIdx.x & 31;
    const int hf   = lane >> 4;
    const int l16  = lane & 15;

    const int wm = wave & 1;             // 2 waves along M
    const int wn = wave >> 1;            // 4 waves along N
    const int M0 = blockIdx.y * 64  + wm * 32;
    const int N0 = blockIdx.x * 128 + wn * 32;

    const float* pa0 = F1 + (size_t)(M0 + l16) * HID + 2 * hf;
    const float* pa1 = pa0 + 16 * HID;
    const float* pb0 = F2 + (size_t)(N0 + l16) * HID + 2 * hf;
    const float* pb1 = pb0 + 16 * HID;

    v8f c00 = {}, c01 = {}, c10 = {}, c11 = {};
    #pragma unroll
    for (int kb = 0; kb < HID; kb += 4) {
        v2f a0 = *(const v2f*)(pa0 + kb);
        v2f a1 = *(const v2f*)(pa1 + kb);
        v2f b0 = *(const v2f*)(pb0 + kb);
        v2f b1 = *(const v2f*)(pb1 + kb);
        c00 = __builtin_amdgcn_wmma_f32_16x16x4_f32(false, a0, false, b0,
                                                    (short)0, c00, false, false);
        c01 = __builtin_amdgcn_wmma_f32_16x16x4_f32(false, a0, false, b1,
                                                    (short)0, c01, false, false);
        c10 = __builtin_amdgcn_wmma_f32_16x16x4_f32(false, a1, false, b0,
                                                    (short)0, c10, false, false);
        c11 = __builtin_amdgcn_wmma_f32_16x16x4_f32(false, a1, false, b1,
                                                    (short)0, c11, false, false);
    }

    const float sn0 = sq2[N0 + l16];         // column for ni=0 tiles
    const float sn1 = sq2[N0 + 16 + l16];    // column for ni=1 tiles

    #pragma unroll
    for (int r = 0; r < 8; ++r) {
        const int m0 = M0 + r + 8 * hf;      // row for mi=0 tiles
        const int m1 = m0 + 16;              // row for mi=1 tiles
        const float sm0 = sq1[m0];
        const float sm1 = sq1[m1];

        float* o0 = out + (size_t)m0 * SP + N0 + l16;
        float* o1 = out + (size_t)m1 * SP + N0 + l16;

        __builtin_nontemporal_store(sig_dist(c00[r], sm0, sn0), o0);
        __builtin_nontemporal_store(sig_dist(c01[r], sm0, sn1), o0 + 16);
        __builtin_nontemporal_store(sig_dist(c10[r], sm1, sn0), o1);
        __builtin_nontemporal_store(sig_dist(c11[r], sm1, sn1), o1 + 16);
    }
}

extern "C" void kernel_launch(void* const* d_in, const int* in_sizes, int n_in,
                              void* d_out, int out_size, void* d_ws, size_t ws_size,
                              hipStream_t stream) {
    const float* input1 = (const float*)d_in[0];   // [8192,256]
    const float* input2 = (const float*)d_in[1];   // [8192,256]
    const float* weight = (const float*)d_in[2];   // [256,64]
    float* out = (float*)d_out;                    // [8192,8192]

    float* F1  = (float*)d_ws;                     // 8192*64
    float* F2  = F1 + (size_t)SP * HID;            // 8192*64
    float* sq1 = F2 + (size_t)SP * HID;            // 8192
    float* sq2 = sq1 + SP;                         // 8192

    // 1) projections + LeakyReLU  (2048 wave-tiles per input, 8 waves/block)
    dim3 pgrid(256, 2);
    proj_wmma_kernel<<<pgrid, 256, 0, stream>>>(input1, input2, weight, F1, F2);

    // 2) row sums of squares
    rowsq_kernel<<<(2 * SP) / 256, 256, 0, stream>>>(F1, F2, sq1, sq2);

    // 3) pairwise distance + sigmoid epilogue
    dim3 dgrid(SP / 128, SP / 64);
    dist_wmma_kernel<<<dgrid, 256, 0, stream>>>(F1, F2, sq1, sq2, out);
}